// QCNN_ZNOTY_Diff_12756052869706
// MI455X (gfx1250) — compile-verified
//
#include <hip/hip_runtime.h>
#include <hip/hip_bf16.h>
#include <math.h>

typedef float v2f __attribute__((ext_vector_type(2)));
typedef float v8f __attribute__((ext_vector_type(8)));

#define NQ 8
#define DIM 256
#define PI_HALF 1.57079632679489662f

// ---------------------------------------------------------------------------
// Phase 1: build the two branch unitaries by simulating the 256 basis states.
// Block b in [0,512): branch = b>>8, basis state k = b&255. 128 threads.
// Output layout (row-major over output index n): U[n*256 + k] = <n|U|k>,
// stored as Ubase + branch*131072 (+65536 for imag). This is exactly the
// B-matrix (k-major per column) the WMMA GEMM consumes.
// ---------------------------------------------------------------------------

__device__ inline void g_apply1(float* sre, float* sim, int w,
                                float u00r, float u00i, float u01r, float u01i,
                                float u10r, float u10i, float u11r, float u11i) {
  __syncthreads();
  int t = threadIdx.x;            // 0..127 -> one amplitude pair each
  int bp = 7 - w;
  int m = 1 << bp;
  int i0 = ((t & ~(m - 1)) << 1) | (t & (m - 1));
  int i1 = i0 | m;
  float a0r = sre[i0], a0i = sim[i0];
  float a1r = sre[i1], a1i = sim[i1];
  sre[i0] = u00r * a0r - u00i * a0i + u01r * a1r - u01i * a1i;
  sim[i0] = u00r * a0i + u00i * a0r + u01r * a1i + u01i * a1r;
  sre[i1] = u10r * a0r - u10i * a0i + u11r * a1r - u11i * a1i;
  sim[i1] = u10r * a0i + u10i * a0r + u11r * a1i + u11i * a1r;
}

__device__ inline void rz_gate(float* sre, float* sim, int w, float t) {
  float ch = cosf(0.5f * t), sh = sinf(0.5f * t);
  g_apply1(sre, sim, w, ch, -sh, 0.f, 0.f, 0.f, 0.f, ch, sh);
}

__device__ inline void ry_gate(float* sre, float* sim, int w, float t) {
  float ch = cosf(0.5f * t), sh = sinf(0.5f * t);
  g_apply1(sre, sim, w, ch, 0.f, -sh, 0.f, sh, 0.f, ch, 0.f);
}

__device__ inline void u3_gate(float* sre, float* sim, int w,
                               float th, float p, float l) {
  float c = cosf(0.5f * th), s = sinf(0.5f * th);
  float cl = cosf(l), sl = sinf(l);
  float cp = cosf(p), sp = sinf(p);
  float cpl = cosf(p + l), spl = sinf(p + l);
  g_apply1(sre, sim, w,
           c, 0.f, -cl * s, -sl * s,
           cp * s, sp * s, cpl * c, spl * c);
}

__device__ inline void quad_idx(int t, int bpc, int bpt, int* pi10, int* pi11) {
  int lo = bpc < bpt ? bpc : bpt;
  int hi = bpc < bpt ? bpt : bpc;
  int ml = 1 << lo, mh = 1 << hi;
  int i = ((t & ~(ml - 1)) << 1) | (t & (ml - 1));
  i = ((i & ~(mh - 1)) << 1) | (i & (mh - 1));
  int i10 = i | (1 << bpc);
  *pi10 = i10;
  *pi11 = i10 | (1 << bpt);
}

__device__ inline void cx_gate(float* sre, float* sim, int cw, int tw) {
  __syncthreads();
  int t = threadIdx.x;
  if (t < 64) {
    int i10, i11;
    quad_idx(t, 7 - cw, 7 - tw, &i10, &i11);
    float r = sre[i10], q = sim[i10];
    sre[i10] = sre[i11]; sim[i10] = sim[i11];
    sre[i11] = r;        sim[i11] = q;
  }
}

__device__ inline void crx_gate(float* sre, float* sim, int cw, int tw, float th) {
  __syncthreads();
  int t = threadIdx.x;
  if (t < 64) {
    int i10, i11;
    quad_idx(t, 7 - cw, 7 - tw, &i10, &i11);
    float c = cosf(0.5f * th), s = sinf(0.5f * th);
    float ar = sre[i10], ai = sim[i10];
    float br = sre[i11], bi = sim[i11];
    sre[i10] = c * ar + s * bi;     // c*a + (-i s)*b
    sim[i10] = c * ai - s * br;
    sre[i11] = s * ai + c * br;     // (-i s)*a + c*b
    sim[i11] = -s * ar + c * bi;
  }
}

__device__ inline void conv_block(float* sre, float* sim, int a, int b,
                                  float rz, float ry, float ry2) {
  rz_gate(sre, sim, b, -PI_HALF);
  cx_gate(sre, sim, b, a);
  rz_gate(sre, sim, a, rz);
  ry_gate(sre, sim, b, ry);
  cx_gate(sre, sim, a, b);
  ry_gate(sre, sim, b, ry2);
  cx_gate(sre, sim, b, a);
  rz_gate(sre, sim, a, PI_HALF);
}

__global__ __launch_bounds__(128) void qcnn_build_unitaries(
    const float* __restrict__ conv_rz, const float* __restrict__ conv_ry,
    const float* __restrict__ conv_ry2, const float* __restrict__ pool_u3,
    const float* __restrict__ crx, const float* __restrict__ u3x,
    float* __restrict__ Ubase) {
  __shared__ float sre[DIM];
  __shared__ float sim[DIM];
  int t = threadIdx.x;
  int br = blockIdx.x >> 8;
  int k = blockIdx.x & 255;
  sre[t] = 0.f; sim[t] = 0.f;
  sre[t + 128] = 0.f; sim[t + 128] = 0.f;
  __syncthreads();
  if (t == 0) sre[k] = 1.f;

  if (br == 0) {
    // QCNN branch: layer 0 convs on pairs, then odd pairs, pool, layer 1.
    const int p0[7][2] = {{0,1},{2,3},{4,5},{6,7},{1,2},{3,4},{5,6}};
    for (int cc = 0; cc < 7; ++cc)
      conv_block(sre, sim, p0[cc][0], p0[cc][1],
                 conv_rz[cc], conv_ry[cc], conv_ry2[cc]);
    const int pq0[4] = {1, 3, 5, 7};
    for (int pc = 0; pc < 4; ++pc)
      u3_gate(sre, sim, pq0[pc],
              pool_u3[pc * 3 + 0], pool_u3[pc * 3 + 1], pool_u3[pc * 3 + 2]);
    const int p1[3][2] = {{1,3},{5,7},{3,5}};
    for (int j = 0; j < 3; ++j)
      conv_block(sre, sim, p1[j][0], p1[j][1],
                 conv_rz[7 + j], conv_ry[7 + j], conv_ry2[7 + j]);
    u3_gate(sre, sim, 3, pool_u3[12], pool_u3[13], pool_u3[14]);
    u3_gate(sre, sim, 7, pool_u3[15], pool_u3[16], pool_u3[17]);
  } else {
    // CRX feature-map branch.
    const int cp0[7][2] = {{0,1},{2,3},{4,5},{6,7},{1,2},{3,4},{5,6}};
    for (int j = 0; j < 7; ++j)
      crx_gate(sre, sim, cp0[j][0], cp0[j][1], crx[j]);
    const int uq[4] = {1, 3, 5, 7};
    for (int j = 0; j < 4; ++j)
      u3_gate(sre, sim, uq[j], u3x[j * 3 + 0], u3x[j * 3 + 1], u3x[j * 3 + 2]);
    const int cp1[3][2] = {{1,3},{5,7},{3,5}};
    for (int j = 0; j < 3; ++j)
      crx_gate(sre, sim, cp1[j][0], cp1[j][1], crx[7 + j]);
    u3_gate(sre, sim, 3, u3x[12], u3x[13], u3x[14]);
    u3_gate(sre, sim, 7, u3x[15], u3x[16], u3x[17]);
  }
  __syncthreads();
  float* Ure = Ubase + br * (2 * DIM * DIM);
  float* Uim = Ure + DIM * DIM;
  Ure[t * DIM + k] = sre[t];               Uim[t * DIM + k] = sim[t];
  Ure[(t + 128) * DIM + k] = sre[t + 128]; Uim[(t + 128) * DIM + k] = sim[t + 128];
}

// ---------------------------------------------------------------------------
// Phase 2: batched complex GEMM via V_WMMA_F32_16X16X4_F32 + feature
// reduction + MLP head. One WG of 16 waves per 16-row batch tile.
// ---------------------------------------------------------------------------

#define PSI_LD 260  // 256 + 4 pad -> bank-conflict-free (260 mod 64 == 4)

__global__ __launch_bounds__(512) void qcnn_wmma_main(
    const float* __restrict__ state_re, const float* __restrict__ state_im,
    const float* __restrict__ Ubase,
    const float* __restrict__ w1, const float* __restrict__ b1,
    const float* __restrict__ w2, const float* __restrict__ b2,
    float* __restrict__ out) {
  __shared__ float lre[16 * PSI_LD];
  __shared__ float lim[16 * PSI_LD];
  __shared__ float featAcc[16][4];

  const int m0 = blockIdx.x * 16;
  const int tid = threadIdx.x;

  // Coalesced stage of the 16x256 psi tile (re & im) into LDS.
  for (int idx = tid; idx < 16 * DIM; idx += 512) {
    int r = idx >> 8, c = idx & 255;
    lre[r * PSI_LD + c] = state_re[(m0 + r) * DIM + c];
    lim[r * PSI_LD + c] = state_im[(m0 + r) * DIM + c];
  }
  if (tid < 64) featAcc[tid >> 2][tid & 3] = 0.f;
  __syncthreads();

  const int wave = tid >> 5;          // 0..15 -> N-tile
  const int lane = tid & 31;
  const int n0 = wave * 16;
  const int row = lane & 15;          // A: M row  /  B: N column (lane&15)
  const int kk = (lane >> 4) << 1;    // ISA 16x4 layout: half-wave K split
  const float* arow_re = &lre[row * PSI_LD];
  const float* arow_im = &lim[row * PSI_LD];

  for (int br = 0; br < 2; ++br) {
    const float* Ure = Ubase + br * (2 * DIM * DIM);
    const float* Uim = Ure + DIM * DIM;
    const float* bre = Ure + (n0 + row) * DIM + kk;  // B[k][n] = U[n][k]
    const float* bim = Uim + (n0 + row) * DIM + kk;

    v8f accR = {0.f, 0.f, 0.f, 0.f, 0.f, 0.f, 0.f, 0.f};
    v8f accI = {0.f, 0.f, 0.f, 0.f, 0.f, 0.f, 0.f, 0.f};

    for (int k0 = 0; k0 < DIM; k0 += 4) {
      v2f aR = *(const v2f*)(arow_re + k0 + kk);
      v2f aI = *(const v2f*)(arow_im + k0 + kk);
      v2f bR = *(const v2f*)(bre + k0);
      v2f bI = *(const v2f*)(bim + k0);
      v2f aIn = -aI;
      // out_re = Re*Re - Im*Im ; out_im = Re*Im + Im*Re
      accR = __builtin_amdgcn_wmma_f32_16x16x4_f32(
          false, aR, false, bR, (short)0, accR, false, false);
      accR = __builtin_amdgcn_wmma_f32_16x16x4_f32(
          false, aIn, false, bI, (short)0, accR, false, false);
      accI = __builtin_amdgcn_wmma_f32_16x16x4_f32(
          false, aR, false, bI, (short)0, accI, false, false);
      accI = __builtin_amdgcn_wmma_f32_16x16x4_f32(
          false, aI, false, bR, (short)0, accI, false, false);
    }

    // Signed |amp|^2 reduction: wire 3 -> bit 4, wire 7 -> bit 0.
    int nc = n0 + (lane & 15);
    float s3 = ((nc >> 4) & 1) ? -1.f : 1.f;
    float s7 = (nc & 1) ? -1.f : 1.f;
    int mbase = (lane >> 4) << 3;  // C/D layout: half-wave -> M +8
    for (int v = 0; v < 8; ++v) {
      float p = accR[v] * accR[v] + accI[v] * accI[v];
      float f3 = s3 * p, f7 = s7 * p;
      for (int d = 1; d < 16; d <<= 1) {   // reduce within each 16-lane half
        f3 += __shfl_xor(f3, d, 32);
        f7 += __shfl_xor(f7, d, 32);
      }
      if ((lane & 15) == 0) {
        int mrow = mbase + v;
        atomicAdd(&featAcc[mrow][br * 2 + 0], f3);
        atomicAdd(&featAcc[mrow][br * 2 + 1], f7);
      }
    }
  }
  __syncthreads();

  // MLP head: tanh(feat @ w1^T + b1) @ w2^T + b2 -> sigmoid.
  if (tid < 16) {
    float f0 = featAcc[tid][0], f1 = featAcc[tid][1];
    float f2 = featAcc[tid][2], f3 = featAcc[tid][3];
    float o = b2[0];
    for (int j = 0; j < 10; ++j) {
      float h = b1[j] + f0 * w1[j * 4 + 0] + f1 * w1[j * 4 + 1] +
                f2 * w1[j * 4 + 2] + f3 * w1[j * 4 + 3];
      o += tanhf(h) * w2[j];
    }
    out[m0 + tid] = 1.f / (1.f + expf(-o));
  }
}

// ---------------------------------------------------------------------------

extern "C" void kernel_launch(void* const* d_in, const int* in_sizes, int n_in,
                              void* d_out, int out_size, void* d_ws, size_t ws_size,
                              hipStream_t stream) {
  const float* state_re = (const float*)d_in[0];
  const float* state_im = (const float*)d_in[1];
  const float* conv_rz  = (const float*)d_in[2];
  const float* conv_ry  = (const float*)d_in[3];
  const float* conv_ry2 = (const float*)d_in[4];
  const float* pool_u3  = (const float*)d_in[5];
  const float* crx      = (const float*)d_in[6];
  const float* u3x      = (const float*)d_in[7];
  const float* w1       = (const float*)d_in[8];
  const float* b1       = (const float*)d_in[9];
  const float* w2       = (const float*)d_in[10];
  const float* b2       = (const float*)d_in[11];

  float* U = (float*)d_ws;  // 4 x 256x256 f32 matrices = 1 MB

  // 512 blocks: 2 branches x 256 basis states.
  qcnn_build_unitaries<<<512, 128, 0, stream>>>(
      conv_rz, conv_ry, conv_ry2, pool_u3, crx, u3x, U);

  int nB = in_sizes[0] / DIM;               // 16384
  int grid = (nB + 15) / 16;                // 1024 workgroups
  qcnn_wmma_main<<<grid, 512, 0, stream>>>(
      state_re, state_im, U, w1, b1, w2, b2, (float*)d_out);
}